// LearnableConvCensus_27522150432939
// MI455X (gfx1250) — compile-verified
//
#include <hip/hip_runtime.h>

// ---------------------------------------------------------------------------
// LearnableConvCensus for MI455X (gfx1250, wave32)
// depthwise conv (C groups, MULT=8) via V_WMMA_F32_16X16X4_F32,
// tile staging via GLOBAL_LOAD_ASYNC_TO_LDS_B32 (ASYNCcnt),
// mean-of-sigmoids via v_tanh_f32.
// ---------------------------------------------------------------------------

#define NB   4
#define NC   64
#define NH   256
#define NW   256
#define MULT 8
#define CM   (NC * MULT)   // 512

#define TDIM 64            // output tile is 64x64
#define TLW  66            // tile + halo
#define LW   68            // padded LDS row stride (floats)

typedef __attribute__((ext_vector_type(2))) float v2f;
typedef __attribute__((ext_vector_type(8))) float v8f;

#if defined(__HIP_DEVICE_COMPILE__) && !__has_builtin(__builtin_amdgcn_wmma_f32_16x16x4_f32)
#error "missing __builtin_amdgcn_wmma_f32_16x16x4_f32 for gfx1250"
#endif

__device__ __forceinline__ float fast_tanh(float x) {
#if defined(__HIP_DEVICE_COMPILE__)
#if __has_builtin(__builtin_amdgcn_tanhf)
    return __builtin_amdgcn_tanhf(x);
#elif __has_builtin(__builtin_amdgcn_tanh_f32)
    return __builtin_amdgcn_tanh_f32(x);
#else
    float e = __builtin_amdgcn_exp2f(x * -2.885390081777927f);
    return __builtin_amdgcn_rcpf(1.0f + e) * 2.0f - 1.0f;
#endif
#else
    return x;  // host pass: never executed
#endif
}

__global__ __launch_bounds__(256)
void census_wmma_kernel(const float* __restrict__ x,
                        const float* __restrict__ w,
                        const float* __restrict__ bias,
                        const float* __restrict__ temp,
                        float* __restrict__ out) {
    __shared__ float tile[TLW * LW];

    const int tid  = threadIdx.x;
    const int lane = tid & 31;
    const int wv   = tid >> 5;          // wave id 0..7
    const bool hi  = lane >= 16;

    const int blk   = blockIdx.x;       // [0, NB*NC*16)
    const int plane = blk >> 4;         // b*NC + c
    const int c     = plane & (NC - 1);
    const int t4    = blk & 15;
    const int x0    = (t4 & 3) * TDIM;
    const int y0    = (t4 >> 2) * TDIM;

    const float* xp = x + (size_t)plane * NH * NW;
    float*       op = out + (size_t)plane * NH * NW;

    // ---- stage 64x64 tile + 1-halo into LDS ----
    // In-range cells: async VMEM->LDS (ASYNCcnt). OOB halo cells: ds_store 0.
    // Each LDS cell is written by exactly one path, so no ordering hazard.
    for (int idx = tid; idx < TLW * TLW; idx += 256) {
        int ly = idx / TLW;
        int lx = idx - ly * TLW;
        int gy = y0 + ly - 1;
        int gx = x0 + lx - 1;
        float* dst = &tile[ly * LW + lx];
        if ((unsigned)gy < (unsigned)NH && (unsigned)gx < (unsigned)NW) {
            const float* src = xp + gy * NW + gx;
            unsigned ldsoff = (unsigned)(size_t)dst;  // low 32b of flat = LDS offset
            asm volatile("global_load_async_to_lds_b32 %0, %1, off"
                         :: "v"(ldsoff), "v"(src) : "memory");
        } else {
            *dst = 0.0f;
        }
    }

    // ---- per-plane constants ----
    // A matrix (16x4 f32): lane holds row M = lane&15 (rows 8..15 duplicate
    // channels 0..7 via nn = lane&7), VGPR0 = K (lo:0 / hi:2), VGPR1 = K+1.
    // K group j maps to kernel row kh=j, taps kw = {0,1 | 2,zero}.
    const int nn = lane & 7;
    const int k0 = hi ? 2 : 0;
    v2f wa[3];
#pragma unroll
    for (int j = 0; j < 3; ++j) {
        float wy = w[(j * 3 + 1) * CM + c * MULT + nn];
        wa[j].x = w[(j * 3 + k0) * CM + c * MULT + nn];
        wa[j].y = hi ? 0.0f : wy;   // hi-lane K=3 weight is zero
    }
    float bv[8];
#pragma unroll
    for (int r = 0; r < 8; ++r) bv[r] = bias[c * MULT + r];
    const float th = 0.5f * temp[c];    // sigmoid(x) = 0.5 + 0.5*tanh(0.5*x)

    asm volatile("s_wait_asynccnt 0" ::: "memory");
    __syncthreads();

    const int m = lane & 15;            // pixel (B-matrix column N)
    // each wave owns 8 consecutive tile rows
    for (int i = 0; i < 8; ++i) {
        const int ry = wv * 8 + i;
        float* orow = op + (size_t)(y0 + ry) * NW + x0;
        for (int g = 0; g < 4; ++g) {
            const int c0  = g * 16;
            const int col = 1 + c0 + m;
            v8f acc;
#pragma unroll
            for (int r = 0; r < 8; ++r) acc[r] = bv[r];
#pragma unroll
            for (int j = 0; j < 3; ++j) {
                // B matrix (4x16 f32): lane holds col N = lane&15,
                // VGPR0 = K (lo:0 / hi:2), VGPR1 = K+1.
                // Hi-lane B[3,N] is garbage but multiplied by A[M,3] == 0.
                const int base = (ry + j) * LW + col;   // row = ry+1 + (j-1)
                v2f bm;
                bm.x = tile[base + (hi ? 1 : -1)];
                bm.y = tile[base];
                acc = __builtin_amdgcn_wmma_f32_16x16x4_f32(
                        false, wa[j], false, bm, (short)0, acc, false, false);
            }
            // D: reg r, lo lanes = channel r; hi lanes = channel r (duplicate
            // A rows). Split the 8 tanh's across the two halves.
            float e0 = hi ? acc[4] : acc[0];
            float e1 = hi ? acc[5] : acc[1];
            float e2 = hi ? acc[6] : acc[2];
            float e3 = hi ? acc[7] : acc[3];
            float p = fast_tanh(e0 * th) + fast_tanh(e1 * th) +
                      fast_tanh(e2 * th) + fast_tanh(e3 * th);
            p += __shfl_xor(p, 16, 32);
            // mean of 8 sigmoids = 0.5 + (1/16) * sum(tanh)
            if (!hi) orow[c0 + m] = fmaf(p, 0.0625f, 0.5f);
        }
    }
}

extern "C" void kernel_launch(void* const* d_in, const int* in_sizes, int n_in,
                              void* d_out, int out_size, void* d_ws, size_t ws_size,
                              hipStream_t stream) {
    (void)in_sizes; (void)n_in; (void)out_size; (void)d_ws; (void)ws_size;
    const float* x = (const float*)d_in[0];
    const float* w = (const float*)d_in[1];
    const float* b = (const float*)d_in[2];
    const float* t = (const float*)d_in[3];
    float* out = (float*)d_out;

    dim3 grid(NB * NC * 16);   // 4096 workgroups, one 64x64 tile each
    dim3 block(256);           // 8 waves (wave32)
    census_wmma_kernel<<<grid, block, 0, stream>>>(x, w, b, t, out);
}